// OnLSTM_88124138979439
// MI455X (gfx1250) — compile-verified
//
#include <hip/hip_runtime.h>

// ---- problem constants (from the reference) ----
constexpr int kH     = 1024;           // hidden size
constexpr int kLH    = 64;             // level hidden
constexpr int kB     = 64;             // batch
constexpr int kT     = 256;            // time
constexpr int kD0    = 512;            // layer-0 input width
constexpr int k4H    = 4 * kH;         // 4096
constexpr int k2LH   = 2 * kLH;        // 128
constexpr int kNG    = k4H + k2LH;     // 4224 fused gate columns

typedef float v2f __attribute__((ext_vector_type(2)));
typedef float v8f __attribute__((ext_vector_type(8)));

// WGP-scope prefetch (SCOPE field = 0): pulls the line into ALL cache levels,
// unlike __builtin_prefetch which only reaches SE/SYS scope (GL2-only).
// NOTE: WGP-scope prefetch is non-speculative -> caller must keep it in-bounds.
__device__ __forceinline__ void prefetch_wgp(const float* p) {
    asm volatile("global_prefetch_b8 %0, off" :: "v"(p));
}

// One k-step (K+=4) for a 16x32 output strip: 1 A fragment feeds 2 WMMAs.
//   A frag (16x4 f32): lane -> (M = lane&15), VGPRs = K = k+2*(lane>>4), +1
//   B frag (4x16 f32): lane -> (N = lane&15), VGPRs = K = k+2*(lane>>4), +1
__device__ __forceinline__ void gemm_kstep(const float* __restrict__ asrc, // + k + 2*half
                                           const float* __restrict__ bptr, // + k*ldB + col
                                           int ldB, int half, v8f* acc)
{
    const v2f a = *(const v2f*)asrc;
    const float* bp = bptr + (size_t)(2 * half) * ldB;
    v2f bf0, bf1;
    bf0.x = bp[0];    bf0.y = bp[ldB];
    bf1.x = bp[16];   bf1.y = bp[ldB + 16];
    acc[0] = __builtin_amdgcn_wmma_f32_16x16x4_f32(false, a, false, bf0,
                                                   (short)0, acc[0], false, false);
    acc[1] = __builtin_amdgcn_wmma_f32_16x16x4_f32(false, a, false, bf1,
                                                   (short)0, acc[1], false, false);
}

// -----------------------------------------------------------------------------
// Per-timestep GEMM:  G[b, n] = sum_k [x_t | h][b, k] * [W | L][k, n] + [bias|lb][n]
// One wave per 16x32 output strip (2 N-tiles), fp32 WMMA 16x16x4 accumulation.
// K range split at compile time: [0,DIN) reads x_t, [DIN,DIN+H) reads h.
// Weight matrices stay L2-resident (25-33 MB << 192 MB L2); WGP-scope prefetch
// warms L0 ~4 k-steps (16 rows) ahead of use. Loop-1 prefetch overshoot lands
// in the h-rows of the same contiguous matrix (valid); loop-2 is guarded.
// -----------------------------------------------------------------------------
template <int DIN>
__global__ __launch_bounds__(32)
void onlstm_step_gemm(const float* __restrict__ xin,   // (B, T, DIN)
                      const float* __restrict__ hbuf,  // (B, H)
                      const float* __restrict__ W,     // (DIN+H, 4H)
                      const float* __restrict__ Lm,    // (DIN+H, 2LH)
                      const float* __restrict__ bias,  // (4H)
                      const float* __restrict__ lbias, // (2LH)
                      float* __restrict__ G,           // (B, NG)
                      int t)
{
    const int lane = threadIdx.x;
    const int half = lane >> 4;      // 0: lanes 0-15, 1: lanes 16-31
    const int l15  = lane & 15;
    const int n0   = blockIdx.x * 32;   // strip of 2 x 16 columns
    const int m0   = blockIdx.y * 16;

    const int brow = m0 + l15;          // batch row this lane loads for A
    const float* xrow = xin + ((size_t)brow * kT + t) * (size_t)DIN + 2 * half;
    const float* hrow = hbuf + (size_t)brow * kH + 2 * half;

    // 4096 % 32 == 0 -> each strip lies entirely in W or entirely in L
    const bool   isW = (n0 < k4H);
    const int    col = (isW ? n0 : n0 - k4H) + l15;
    const int    ldB = isW ? k4H : k2LH;
    const float* bptr = (isW ? W : Lm) + col;

    v8f acc[2] = {};

#pragma unroll 4
    for (int k = 0; k < DIN; k += 4) {
        prefetch_wgp(bptr + (size_t)16 * ldB);    // rows k+16..k+19 < DIN+kH: in-bounds
        gemm_kstep(xrow + k, bptr, ldB, half, acc);
        bptr += (size_t)4 * ldB;
    }
#pragma unroll 4
    for (int k = 0; k < kH; k += 4) {
        if (k < kH - 20)                          // keep non-speculative prefetch in-bounds
            prefetch_wgp(bptr + (size_t)16 * ldB);
        gemm_kstep(hrow + k, bptr, ldB, half, acc);
        bptr += (size_t)4 * ldB;
    }

    const float* bvec = isW ? bias : lbias;
    const float bv0 = bvec[col];
    const float bv1 = bvec[col + 16];
#pragma unroll
    for (int r = 0; r < 8; ++r) {
        const int mrow = m0 + r + 8 * half;     // C/D layout: VGPR r -> row r+8*half
        float* gr = G + (size_t)mrow * kNG + n0 + l15;
        gr[0]  = acc[0][r] + bv0;
        gr[16] = acc[1][r] + bv1;
    }
}

// -----------------------------------------------------------------------------
// Per-timestep cell update: softmax+cumsum over level logits, ON-LSTM gating.
// One block per batch row.
// -----------------------------------------------------------------------------
__global__ __launch_bounds__(256)
void onlstm_step_gates(const float* __restrict__ G,    // (B, NG)
                       float* __restrict__ hbuf,       // (B, H) in/out
                       float* __restrict__ cbuf,       // (B, H) in/out
                       float* __restrict__ ys,         // (B, T, H)
                       int t)
{
    __shared__ float s_pf[kLH];
    __shared__ float s_pi[kLH];
    __shared__ float s_fh[kLH];
    __shared__ float s_ih[kLH];

    const int b   = blockIdx.x;
    const int tid = threadIdx.x;
    const float* g = G + (size_t)b * kNG;

    if (tid < kLH) {
        s_pi[tid] = g[k4H + tid];           // cc_ih  (cols 4096..4159)
        s_pf[tid] = g[k4H + kLH + tid];     // cc_fh  (cols 4160..4223)
    }
    __syncthreads();

    if (tid == 0) {                          // master forget: fwd cumsum of softmax
        float mx = s_pf[0];
        for (int j = 1; j < kLH; ++j) mx = fmaxf(mx, s_pf[j]);
        float sum = 0.f;
        for (int j = 0; j < kLH; ++j) { float e = __expf(s_pf[j] - mx); s_pf[j] = e; sum += e; }
        const float inv = 1.f / sum;
        float run = 0.f;
        for (int j = 0; j < kLH; ++j) { run += s_pf[j] * inv; s_fh[j] = run; }
    } else if (tid == 32) {                  // master input: reverse cumsum of softmax
        float mx = s_pi[0];
        for (int j = 1; j < kLH; ++j) mx = fmaxf(mx, s_pi[j]);
        float sum = 0.f;
        for (int j = 0; j < kLH; ++j) { float e = __expf(s_pi[j] - mx); s_pi[j] = e; sum += e; }
        const float inv = 1.f / sum;
        float run = 0.f;
        for (int j = kLH - 1; j >= 0; --j) { run += s_pi[j] * inv; s_ih[j] = run; }
    }
    __syncthreads();

    for (int idx = tid; idx < kH; idx += 256) {
        const int lev = idx >> 4;            // idx / (H/LH)
        const float iv = 1.f / (1.f + __expf(-g[idx]));
        const float fv = 1.f / (1.f + __expf(-g[kH + idx]));
        const float ov = 1.f / (1.f + __expf(-g[2 * kH + idx]));
        const float gv = tanhf(g[3 * kH + idx]);
        const float fh = s_fh[lev];
        const float ih = s_ih[lev];
        const float w  = ih * fh;
        const size_t o = (size_t)b * kH + idx;
        const float cold = cbuf[o];
        const float cnew = w * (fv * cold + iv * gv) + (fh - w) * cold + (ih - w) * gv;
        const float hnew = ov * tanhf(cnew);
        cbuf[o] = cnew;
        hbuf[o] = hnew;
        ys[((size_t)b * kT + t) * kH + idx] = hnew;
    }
}

__global__ void zero2_kernel(float* __restrict__ a, float* __restrict__ b, int n) {
    int i = blockIdx.x * blockDim.x + threadIdx.x;
    if (i < n) { a[i] = 0.f; b[i] = 0.f; }
}

__global__ void copy_hc_kernel(const float* __restrict__ h, const float* __restrict__ c,
                               float* __restrict__ oh, float* __restrict__ oc, int n) {
    int i = blockIdx.x * blockDim.x + threadIdx.x;
    if (i < n) { oh[i] = h[i]; oc[i] = c[i]; }
}

extern "C" void kernel_launch(void* const* d_in, const int* in_sizes, int n_in,
                              void* d_out, int out_size, void* d_ws, size_t ws_size,
                              hipStream_t stream)
{
    const float* x   = (const float*)d_in[0];
    const float* W0  = (const float*)d_in[1];
    const float* b0  = (const float*)d_in[2];
    const float* L0  = (const float*)d_in[3];
    const float* lb0 = (const float*)d_in[4];
    const float* W1  = (const float*)d_in[5];
    const float* b1  = (const float*)d_in[6];
    const float* L1  = (const float*)d_in[7];
    const float* lb1 = (const float*)d_in[8];

    float* out  = (float*)d_out;
    float* y1   = out;                                   // (B, T, H)
    float* outh = out + (size_t)kB * kT * kH;            // (2, B, H)
    float* outc = outh + (size_t)2 * kB * kH;            // (2, B, H)

    // workspace layout
    float* ws   = (float*)d_ws;
    float* G    = ws;                                    // B * NG
    float* hbuf = G + (size_t)kB * kNG;                  // B * H
    float* cbuf = hbuf + (size_t)kB * kH;                // B * H
    float* y0   = cbuf + (size_t)kB * kH;                // B * T * H

    const dim3 ggrid(kNG / 32, kB / 16);                 // 132 x 4 strips
    const int  nHC  = kB * kH;
    const int  cblk = (nHC + 255) / 256;

    // ---- layer 0 ----
    zero2_kernel<<<cblk, 256, 0, stream>>>(hbuf, cbuf, nHC);
    for (int t = 0; t < kT; ++t) {
        onlstm_step_gemm<kD0><<<ggrid, 32, 0, stream>>>(x, hbuf, W0, L0, b0, lb0, G, t);
        onlstm_step_gates<<<kB, 256, 0, stream>>>(G, hbuf, cbuf, y0, t);
    }
    copy_hc_kernel<<<cblk, 256, 0, stream>>>(hbuf, cbuf, outh, outc, nHC);

    // ---- layer 1 ----
    zero2_kernel<<<cblk, 256, 0, stream>>>(hbuf, cbuf, nHC);
    for (int t = 0; t < kT; ++t) {
        onlstm_step_gemm<kH><<<ggrid, 32, 0, stream>>>(y0, hbuf, W1, L1, b1, lb1, G, t);
        onlstm_step_gates<<<kB, 256, 0, stream>>>(G, hbuf, cbuf, y1, t);
    }
    copy_hc_kernel<<<cblk, 256, 0, stream>>>(hbuf, cbuf, outh + nHC, outc + nHC, nHC);
}